// gate_GTN_7911329759773
// MI455X (gfx1250) — compile-verified
//
#include <hip/hip_runtime.h>
#include <hip/hip_bf16.h>

#define NN   50000
#define FIN  128
#define DD   64
#define HH   4
#define EDIM 32
#define EE   800000
#define HD   256   // H*D

typedef float v2f __attribute__((ext_vector_type(2)));
typedef float v8f __attribute__((ext_vector_type(8)));

// ---- order-preserving float<->uint map for atomicMax on signed floats ----
__device__ __forceinline__ unsigned f2ord(float f) {
    unsigned u = __float_as_uint(f);
    return (u & 0x80000000u) ? ~u : (u | 0x80000000u);
}
__device__ __forceinline__ float ord2f(unsigned o) {
    unsigned u = (o & 0x80000000u) ? (o & 0x7fffffffu) : ~o;
    return __uint_as_float(u);
}

// ---------------------------------------------------------------------------
// C[M,Nout] = A[M,K] @ W[Nout,K]^T + bias.  One wave computes one 16x16 tile
// with V_WMMA_F32_16X16X4_F32 accumulating over K in steps of 4.
// A 16x4 layout: lanes 0-15 -> M=lane, {K=0,K=1}; lanes 16-31 -> {K=2,K=3}.
// B  4x16 layout mirrors A (K striped the same way, N=lane&15).
// C/D: VGPR r -> M = r (+8 for lanes 16-31), N = lane&15.
// ---------------------------------------------------------------------------
__global__ __launch_bounds__(32) void wmma_gemm_bias(
    const float* __restrict__ A, const float* __restrict__ W,
    const float* __restrict__ bias, float* __restrict__ C,
    int M, int K, int Nout)
{
    const int lane = threadIdx.x & 31;
    const int m0 = blockIdx.y * 16;
    const int n0 = blockIdx.x * 16;
    const int arow = m0 + (lane & 15);
    const int bcol = n0 + (lane & 15);
    const int koff = (lane >> 4) * 2;

    v8f acc = {0.f, 0.f, 0.f, 0.f, 0.f, 0.f, 0.f, 0.f};
    for (int k0 = 0; k0 < K; k0 += 4) {
        v2f a, b;
        a[0] = A[(size_t)arow * K + k0 + koff];
        a[1] = A[(size_t)arow * K + k0 + koff + 1];
        b[0] = W[(size_t)bcol * K + k0 + koff];
        b[1] = W[(size_t)bcol * K + k0 + koff + 1];
        acc = __builtin_amdgcn_wmma_f32_16x16x4_f32(
            false, a, false, b, (short)0, acc, false, false);
    }
    const int rbase = (lane >> 4) * 8;
    const int col = n0 + (lane & 15);
    const float bb = bias[col];
#pragma unroll
    for (int i = 0; i < 8; ++i)
        C[(size_t)(m0 + rbase + i) * Nout + col] = acc[i] + bb;
}

// ---------------------------------------------------------------------------
// Zero the accumulators: agg[N*D], denom[N*H], m_ord[N*H]
// ---------------------------------------------------------------------------
__global__ void init_accum(float* __restrict__ agg, float* __restrict__ denom,
                           unsigned* __restrict__ m_ord)
{
    int idx = blockIdx.x * blockDim.x + threadIdx.x;
    if (idx < NN * DD) agg[idx] = 0.f;
    if (idx < NN * HH) { denom[idx] = 0.f; m_ord[idx] = 0u; }  // ord 0 == -max
}

// ---------------------------------------------------------------------------
// Pass 1: one wave per edge.  Fused edge-linear (recompute e on the fly),
// per-head score = q[dst] . (k[src]+e) / 8, segment max via ordered atomicMax.
// Lane handles 8 contiguous channels (lane*8..+7) => 8 lanes per head.
// ---------------------------------------------------------------------------
__global__ __launch_bounds__(256) void edge_score(
    const float* __restrict__ ea, const float* __restrict__ we,
    const float* __restrict__ be, const float* __restrict__ q,
    const float* __restrict__ k, const int* __restrict__ ei,
    float* __restrict__ score, unsigned* __restrict__ m_ord)
{
    int wave = (blockIdx.x * blockDim.x + threadIdx.x) >> 5;
    int lane = threadIdx.x & 31;
    if (wave >= EE) return;
    int src = ei[wave];
    int dst = ei[EE + wave];

    float eaj = ea[(size_t)wave * EDIM + lane];   // 1 elem per lane, bcast by shfl
    int c0 = lane * 8;
    float partial = 0.f;
#pragma unroll
    for (int i = 0; i < 8; ++i) {
        int c = c0 + i;
        float ec = be[c];
        const float* wr = we + (size_t)c * EDIM;
#pragma unroll
        for (int j = 0; j < EDIM; ++j)
            ec += wr[j] * __shfl(eaj, j, 32);
        float ke = k[(size_t)src * HD + c] + ec;
        partial += q[(size_t)dst * HD + c] * ke;
    }
    // reduce within each group of 8 lanes (one head)
    partial += __shfl_xor(partial, 1, 32);
    partial += __shfl_xor(partial, 2, 32);
    partial += __shfl_xor(partial, 4, 32);
    if ((lane & 7) == 0) {
        int h = lane >> 3;
        float s = partial * 0.125f;               // 1/sqrt(64)
        score[(size_t)wave * HH + h] = s;
        atomicMax(&m_ord[(size_t)dst * HH + h], f2ord(s));
    }
}

// ---------------------------------------------------------------------------
// Pass 2: a = exp(score - m[dst]); overwrite score with a; denom += a
// ---------------------------------------------------------------------------
__global__ void edge_expsum(const int* __restrict__ ei,
                            const unsigned* __restrict__ m_ord,
                            float* __restrict__ score,
                            float* __restrict__ denom)
{
    int idx = blockIdx.x * blockDim.x + threadIdx.x;
    if (idx >= EE * HH) return;
    int e = idx >> 2, h = idx & 3;
    int dst = ei[EE + e];
    float m = ord2f(m_ord[(size_t)dst * HH + h]);
    float a = __expf(score[idx] - m);
    score[idx] = a;
    atomicAdd(&denom[(size_t)dst * HH + h], a);
}

// ---------------------------------------------------------------------------
// Pass 3: one wave per edge.  alpha = a/(denom+eps); msg = (v[src]+e)*alpha;
// reduce the 4 heads across lanes (shfl_down 16,8) then atomicAdd the head
// MEAN (x0.25) into agg[N,D].  64 atomics per edge instead of 256.
// ---------------------------------------------------------------------------
__global__ __launch_bounds__(256) void edge_message(
    const float* __restrict__ ea, const float* __restrict__ we,
    const float* __restrict__ be, const float* __restrict__ v,
    const int* __restrict__ ei, const float* __restrict__ score,
    const float* __restrict__ denom, float* __restrict__ agg)
{
    int wave = (blockIdx.x * blockDim.x + threadIdx.x) >> 5;
    int lane = threadIdx.x & 31;
    if (wave >= EE) return;
    int src = ei[wave];
    int dst = ei[EE + wave];
    int h = lane >> 3;

    float a = score[(size_t)wave * HH + h];
    float alpha = a / (denom[(size_t)dst * HH + h] + 1e-16f);

    float eaj = ea[(size_t)wave * EDIM + lane];
    int c0 = lane * 8;
    float msg[8];
#pragma unroll
    for (int i = 0; i < 8; ++i) {
        int c = c0 + i;
        float ec = be[c];
        const float* wr = we + (size_t)c * EDIM;
#pragma unroll
        for (int j = 0; j < EDIM; ++j)
            ec += wr[j] * __shfl(eaj, j, 32);
        msg[i] = (v[(size_t)src * HD + c] + ec) * alpha;
    }
    // sum heads: lane l (<8) accumulates lanes l, l+8, l+16, l+24
#pragma unroll
    for (int i = 0; i < 8; ++i) {
        float t = msg[i];
        t += __shfl_down(t, 16, 32);
        t += __shfl_down(t, 8, 32);
        if (lane < 8)
            atomicAdd(&agg[(size_t)dst * DD + lane * 8 + i], t * 0.25f);
    }
}

// ---------------------------------------------------------------------------
// Final: one wave per node (2 channels/lane).  x1 = agg + skip, leaky relu,
// sigmoid gate over concat(x0,x2), blend.
// ---------------------------------------------------------------------------
__global__ __launch_bounds__(256) void node_finalize(
    const float* __restrict__ x0, const float* __restrict__ agg,
    const float* __restrict__ skip, const float* __restrict__ wg,
    const float* __restrict__ bg, float* __restrict__ out)
{
    int wave = (blockIdx.x * blockDim.x + threadIdx.x) >> 5;
    int lane = threadIdx.x & 31;
    if (wave >= NN) return;
    size_t base = (size_t)wave * DD;
    int ca = lane, cb = lane + 32;

    float x0a = x0[base + ca], x0b = x0[base + cb];
    float x1a = agg[base + ca] + skip[base + ca];
    float x1b = agg[base + cb] + skip[base + cb];
    float x2a = x1a > 0.f ? x1a : 0.01f * x1a;
    float x2b = x1b > 0.f ? x1b : 0.01f * x1b;

    // wg: [1, 2*D] -> first 64 weight x0, next 64 weight x2
    float p = wg[ca] * x0a + wg[cb] * x0b + wg[DD + ca] * x2a + wg[DD + cb] * x2b;
    p += __shfl_xor(p, 1, 32);
    p += __shfl_xor(p, 2, 32);
    p += __shfl_xor(p, 4, 32);
    p += __shfl_xor(p, 8, 32);
    p += __shfl_xor(p, 16, 32);
    float g = 1.f / (1.f + __expf(-(p + bg[0])));

    out[base + ca] = g * x0a + (1.f - g) * x2a;
    out[base + cb] = g * x0b + (1.f - g) * x2b;
}

// ---------------------------------------------------------------------------
extern "C" void kernel_launch(void* const* d_in, const int* in_sizes, int n_in,
                              void* d_out, int out_size, void* d_ws, size_t ws_size,
                              hipStream_t stream)
{
    const float* x     = (const float*)d_in[0];
    const float* ea    = (const float*)d_in[1];
    const float* fc_w  = (const float*)d_in[2];
    const float* fc_b  = (const float*)d_in[3];
    const float* wq    = (const float*)d_in[4];
    const float* bq    = (const float*)d_in[5];
    const float* wk    = (const float*)d_in[6];
    const float* bk    = (const float*)d_in[7];
    const float* wv    = (const float*)d_in[8];
    const float* bv    = (const float*)d_in[9];
    const float* we    = (const float*)d_in[10];
    const float* be    = (const float*)d_in[11];
    const float* wskip = (const float*)d_in[12];
    const float* bskip = (const float*)d_in[13];
    const float* wg    = (const float*)d_in[14];
    const float* bg    = (const float*)d_in[15];
    const int*   ei    = (const int*)d_in[16];
    float* out = (float*)d_out;

    // workspace layout (floats) — q/k/v kept resident: 154 MB, fits 192 MB L2
    float* ws    = (float*)d_ws;
    float* x0    = ws;                       // N*D
    float* q     = x0    + (size_t)NN * DD;  // N*HD
    float* k     = q     + (size_t)NN * HD;  // N*HD
    float* v     = k     + (size_t)NN * HD;  // N*HD
    float* skip  = v     + (size_t)NN * HD;  // N*D
    float* score = skip  + (size_t)NN * DD;  // E*H
    float* denom = score + (size_t)EE * HH;  // N*H
    float* agg   = denom + (size_t)NN * HH;  // N*D
    unsigned* m_ord = (unsigned*)(agg + (size_t)NN * DD);  // N*H

    dim3 blk32(32);
    // x0 = fc(x):  M=N, K=128, Nout=64
    wmma_gemm_bias<<<dim3(DD / 16, NN / 16), blk32, 0, stream>>>(x, fc_w, fc_b, x0, NN, FIN, DD);
    // q,k,v = linear(x0):  M=N, K=64, Nout=256
    wmma_gemm_bias<<<dim3(HD / 16, NN / 16), blk32, 0, stream>>>(x0, wq, bq, q, NN, DD, HD);
    wmma_gemm_bias<<<dim3(HD / 16, NN / 16), blk32, 0, stream>>>(x0, wk, bk, k, NN, DD, HD);
    wmma_gemm_bias<<<dim3(HD / 16, NN / 16), blk32, 0, stream>>>(x0, wv, bv, v, NN, DD, HD);
    // skip = lin_skip(x0):  M=N, K=64, Nout=64
    wmma_gemm_bias<<<dim3(DD / 16, NN / 16), blk32, 0, stream>>>(x0, wskip, bskip, skip, NN, DD, DD);

    init_accum<<<(NN * DD + 255) / 256, 256, 0, stream>>>(agg, denom, m_ord);

    // edge passes: 8 waves per 256-thread block, one wave per edge
    edge_score<<<EE / 8, 256, 0, stream>>>(ea, we, be, q, k, ei, score, m_ord);
    edge_expsum<<<(EE * HH + 255) / 256, 256, 0, stream>>>(ei, m_ord, score, denom);
    edge_message<<<EE / 8, 256, 0, stream>>>(ea, we, be, v, ei, score, denom, agg);

    node_finalize<<<(NN + 7) / 8, 256, 0, stream>>>(x0, agg, skip, wg, bg, out);
}